// FullNN_66846870995642
// MI455X (gfx1250) — compile-verified
//
#include <hip/hip_runtime.h>
#include <math.h>

// ---------------- problem constants (match reference) ----------------
constexpr int  EP    = 3;        // elements
constexpr int  QEc   = 60000;    // atoms per element
constexpr int  PP    = 40;       // fp length
constexpr int  HH    = 20;       // hidden
constexpr int  NIMG  = 1800;     // images
constexpr int  Qc    = EP * QEc; // 180000 atoms
constexpr int  NB    = NIMG;     // sort bins
constexpr int  CH    = 256;      // atoms per sort chunk
constexpr int  NCHUNK = (Qc + CH - 1) / CH;  // 704

typedef __attribute__((ext_vector_type(2))) float v2f;
typedef __attribute__((ext_vector_type(8))) float v8f;

// LDS layout for mlp_kernel (float offsets); strides chosen bank-conflict-free
constexpr int SL1 = 44;                 // Bl1 stride (40 k + pad)
constexpr int SHB = 36;                 // h-buffer stride (32 n + pad)
constexpr int OFF_BL1 = 0;              // [32][44] W1 col-major, zero-padded
constexpr int OFF_BL2 = OFF_BL1 + 32 * SL1;   // [32][20] W2 col-major
constexpr int OFF_BB2 = OFF_BL2 + 32 * 20;    // [32][20] W2^T col-major == W2 row-major padded
constexpr int OFF_BB1 = OFF_BB2 + 32 * 20;    // [48][20] W1^T col-major == W1 row-major padded
constexpr int OFF_SB1 = OFF_BB1 + 48 * 20;    // [32] b1 padded
constexpr int OFF_SB2 = OFF_SB1 + 32;         // [32] b2 padded
constexpr int OFF_SW3 = OFF_SB2 + 32;         // [32] W3 padded
constexpr int OFF_H   = OFF_SW3 + 32;         // 2 waves x 3 x [16][36]
constexpr int SMEM_FLOATS = OFF_H + 2 * 3 * 16 * SHB;   // 7200 floats = 28.8KB

// =====================================================================
// Sort phase A: per-chunk histograms (deterministic integer atomics)
// =====================================================================
__global__ __launch_bounds__(256) void chunk_hist_kernel(
    const int* __restrict__ imgIdx, int* __restrict__ chunkHist) {
  __shared__ int h[NB];
  const int c = blockIdx.x;
  for (int i = threadIdx.x; i < NB; i += 256) h[i] = 0;
  __syncthreads();
  const int i = c * CH + threadIdx.x;
  if (i < Qc) atomicAdd(&h[imgIdx[i]], 1);
  __syncthreads();
  for (int j = threadIdx.x; j < NB; j += 256)
    chunkHist[(size_t)c * NB + j] = h[j];
}

// =====================================================================
// Sort phase B: per-bin exclusive prefix over chunks (in place) + totals
// =====================================================================
__global__ __launch_bounds__(256) void col_prefix_kernel(
    int* __restrict__ chunkHist, int* __restrict__ counts) {
  const int b = blockIdx.x * 256 + threadIdx.x;
  if (b >= NB) return;
  int run = 0;
  for (int c = 0; c < NCHUNK; ++c) {
    const size_t idx = (size_t)c * NB + b;
    const int t = chunkHist[idx];
    chunkHist[idx] = run;
    run += t;
  }
  counts[b] = run;
}

// =====================================================================
// Sort phase C: exclusive scan of bin totals -> start[NB+1]
// =====================================================================
__global__ __launch_bounds__(256) void scan_kernel(
    const int* __restrict__ counts, int* __restrict__ start) {
  __shared__ int s[256];
  const int tid = threadIdx.x;
  const int base = tid * 8;
  int loc[8];
  int sum = 0;
#pragma unroll
  for (int j = 0; j < 8; ++j) {
    const int b = base + j;
    const int v = (b < NB) ? counts[b] : 0;
    loc[j] = sum;
    sum += v;
  }
  s[tid] = sum;
  __syncthreads();
  for (int off = 1; off < 256; off <<= 1) {
    const int v = s[tid];
    const int add = (tid >= off) ? s[tid - off] : 0;
    __syncthreads();
    s[tid] = v + add;
    __syncthreads();
  }
  const int excl = (tid == 0) ? 0 : s[tid - 1];
#pragma unroll
  for (int j = 0; j < 8; ++j) {
    const int b = base + j;
    if (b < NB) start[b] = excl + loc[j];
  }
  if (tid == 255) start[NB] = s[255];  // == Q
}

// =====================================================================
// Sort phase D: stable in-chunk ranks -> pos[i] (orig atom -> sorted slot)
// =====================================================================
__global__ __launch_bounds__(256) void rank_kernel(
    const int* __restrict__ imgIdx, const int* __restrict__ chunkHist,
    const int* __restrict__ start, int* __restrict__ pos) {
  __shared__ int bins[CH];
  const int c = blockIdx.x;
  const int t = threadIdx.x;
  const int i = c * CH + t;
  int myb = -1;
  if (i < Qc) myb = imgIdx[i];
  bins[t] = myb;
  __syncthreads();
  if (i < Qc) {
    int r = 0;
    for (int j = 0; j < t; ++j) r += (bins[j] == myb) ? 1 : 0;
    pos[i] = start[myb] + chunkHist[(size_t)c * NB + myb] + r;
  }
}

// =====================================================================
// MLP fwd+bwd with fp32 WMMA (V_WMMA_F32_16X16X4_F32).
// Block = 64 threads = 2 waves; each wave owns 16 atoms; one element/block.
// All B-fragments come from zero-padded LDS tiles -> no exec diamonds
// between WMMAs.
// =====================================================================
__global__ __launch_bounds__(64) void mlp_kernel(
    const float* __restrict__ fps, const float* __restrict__ W1,
    const float* __restrict__ b1, const float* __restrict__ W2,
    const float* __restrict__ b2, const float* __restrict__ W3,
    const float* __restrict__ b3, const int* __restrict__ pos,
    float* __restrict__ atomE, float* __restrict__ dE) {
  __shared__ float smem[SMEM_FLOATS];
  const int tid  = threadIdx.x;
  const int wave = tid >> 5;
  const int lane = tid & 31;
  const int l16  = lane & 15;
  const int hi   = lane >> 4;          // 0 or 1
  const int koff = hi ? 2 : 0;
  const int mof  = hi ? 8 : 0;
  const int atomBase = blockIdx.x * 32 + wave * 16;  // 60000 % 32 == 0
  const int e = (blockIdx.x * 32) / QEc;             // one element per block

  const float* __restrict__ W1e = W1 + (size_t)e * PP * HH;
  const float* __restrict__ b1e = b1 + (size_t)e * HH;
  const float* __restrict__ W2e = W2 + (size_t)e * HH * HH;
  const float* __restrict__ b2e = b2 + (size_t)e * HH;
  const float* __restrict__ W3e = W3 + (size_t)e * HH;
  const float  b3e = b3[e];

  // -------- stage zero-padded, column-major B tiles into LDS ----------
  for (int i = tid; i < 32 * SL1; i += 64) {          // 1408 = 64*22
    const int n = i / SL1, k = i % SL1;
    smem[OFF_BL1 + i] = (n < HH && k < PP) ? W1e[k * HH + n] : 0.0f;
  }
  for (int i = tid; i < 32 * 20; i += 64) {           // 640 = 64*10
    const int n = i / 20, k = i % 20;
    smem[OFF_BL2 + i] = (n < HH) ? W2e[k * HH + n] : 0.0f;
  }
  for (int i = tid; i < 32 * 20; i += 64) {           // W2^T col-major
    const int n = i / 20, k = i % 20;
    smem[OFF_BB2 + i] = (n < HH) ? W2e[n * HH + k] : 0.0f;
  }
  for (int i = tid; i < 48 * 20; i += 64) {           // 960 = 64*15, W1^T
    const int n = i / 20, k = i % 20;
    smem[OFF_BB1 + i] = (n < PP) ? W1e[n * HH + k] : 0.0f;
  }
  if (tid < 32) {
    smem[OFF_SB1 + tid] = (tid < HH) ? b1e[tid] : 0.0f;
    smem[OFF_SB2 + tid] = (tid < HH) ? b2e[tid] : 0.0f;
    smem[OFF_SW3 + tid] = (tid < HH) ? W3e[tid] : 0.0f;
  }
  __syncthreads();

  const float* __restrict__ Bl1 = smem + OFF_BL1;
  const float* __restrict__ Bl2 = smem + OFF_BL2;
  const float* __restrict__ Bb2 = smem + OFF_BB2;
  const float* __restrict__ Bb1 = smem + OFF_BB1;
  const float* __restrict__ sB1 = smem + OFF_SB1;
  const float* __restrict__ sB2 = smem + OFF_SB2;
  const float* __restrict__ sW3 = smem + OFF_SW3;
  float* __restrict__ h1 = smem + OFF_H + wave * 3 * 16 * SHB;
  float* __restrict__ h2 = h1 + 16 * SHB;
  float* __restrict__ g  = h2 + 16 * SHB;

  // ---------------- layer 1: z1 = X[16x40] @ W1[40x20] ----------------
  v8f c1a = {};
  v8f c1b = {};
  const float* __restrict__ xrow = fps + ((size_t)atomBase + l16) * PP;
#pragma unroll
  for (int k0 = 0; k0 < PP; k0 += 4) {
    v2f a, b;
    a.x = xrow[k0 + koff];
    a.y = xrow[k0 + koff + 1];
    b.x = Bl1[l16 * SL1 + k0 + koff];
    b.y = Bl1[l16 * SL1 + k0 + koff + 1];
    c1a = __builtin_amdgcn_wmma_f32_16x16x4_f32(false, a, false, b, (short)0,
                                                c1a, false, false);
    b.x = Bl1[(16 + l16) * SL1 + k0 + koff];
    b.y = Bl1[(16 + l16) * SL1 + k0 + koff + 1];
    c1b = __builtin_amdgcn_wmma_f32_16x16x4_f32(false, a, false, b, (short)0,
                                                c1b, false, false);
  }
#pragma unroll
  for (int v = 0; v < 8; ++v) {
    const int m = v + mof;
    h1[m * SHB + l16]      = tanhf(c1a[v] + sB1[l16]);
    h1[m * SHB + 16 + l16] = tanhf(c1b[v] + sB1[16 + l16]);  // pad -> tanh(0)
  }

  // ---------------- layer 2: z2 = H1[16x20] @ W2[20x20] ---------------
  v8f c2a = {};
  v8f c2b = {};
#pragma unroll
  for (int k0 = 0; k0 < HH; k0 += 4) {
    v2f a, b;
    a.x = h1[l16 * SHB + k0 + koff];
    a.y = h1[l16 * SHB + k0 + koff + 1];
    b.x = Bl2[l16 * 20 + k0 + koff];
    b.y = Bl2[l16 * 20 + k0 + koff + 1];
    c2a = __builtin_amdgcn_wmma_f32_16x16x4_f32(false, a, false, b, (short)0,
                                                c2a, false, false);
    b.x = Bl2[(16 + l16) * 20 + k0 + koff];
    b.y = Bl2[(16 + l16) * 20 + k0 + koff + 1];
    c2b = __builtin_amdgcn_wmma_f32_16x16x4_f32(false, a, false, b, (short)0,
                                                c2b, false, false);
  }
#pragma unroll
  for (int v = 0; v < 8; ++v) {
    const int m = v + mof;
    h2[m * SHB + l16]      = tanhf(c2a[v] + sB2[l16]);
    h2[m * SHB + 16 + l16] = tanhf(c2b[v] + sB2[16 + l16]);
  }

  // ------------- layer 3 (energy) + g2 = W3 * (1 - h2^2) --------------
  if (lane < 16) {
    float s = b3e;
    for (int h = 0; h < HH; ++h) s += h2[lane * SHB + h] * sW3[h];
    atomE[pos[atomBase + lane]] = s;
  }
#pragma unroll
  for (int m = 0; m < 16; ++m) {      // each lane owns column n = lane
    const float t = h2[m * SHB + lane];
    g[m * SHB + lane] = sW3[lane] * (1.0f - t * t);   // pad cols -> 0
  }

  // -------- backward L2: g1 = (g2 @ W2^T) * (1 - h1^2) ----------------
  v8f c3a = {};
  v8f c3b = {};
#pragma unroll
  for (int k0 = 0; k0 < HH; k0 += 4) {
    v2f a, b;
    a.x = g[l16 * SHB + k0 + koff];
    a.y = g[l16 * SHB + k0 + koff + 1];
    b.x = Bb2[l16 * 20 + k0 + koff];
    b.y = Bb2[l16 * 20 + k0 + koff + 1];
    c3a = __builtin_amdgcn_wmma_f32_16x16x4_f32(false, a, false, b, (short)0,
                                                c3a, false, false);
    b.x = Bb2[(16 + l16) * 20 + k0 + koff];
    b.y = Bb2[(16 + l16) * 20 + k0 + koff + 1];
    c3b = __builtin_amdgcn_wmma_f32_16x16x4_f32(false, a, false, b, (short)0,
                                                c3b, false, false);
  }
#pragma unroll
  for (int v = 0; v < 8; ++v) {
    const int m = v + mof;
    const float t0 = h1[m * SHB + l16];
    g[m * SHB + l16] = c3a[v] * (1.0f - t0 * t0);
    const float t1 = h1[m * SHB + 16 + l16];
    g[m * SHB + 16 + l16] = c3b[v] * (1.0f - t1 * t1);  // pad -> 0*(1-0)=0
  }

  // ------------- backward L1: dx = g1[16x20] @ W1^T[20x40] ------------
  v8f c4a = {};
  v8f c4b = {};
  v8f c4c = {};
#pragma unroll
  for (int k0 = 0; k0 < HH; k0 += 4) {
    v2f a, b;
    a.x = g[l16 * SHB + k0 + koff];
    a.y = g[l16 * SHB + k0 + koff + 1];
    b.x = Bb1[l16 * 20 + k0 + koff];
    b.y = Bb1[l16 * 20 + k0 + koff + 1];
    c4a = __builtin_amdgcn_wmma_f32_16x16x4_f32(false, a, false, b, (short)0,
                                                c4a, false, false);
    b.x = Bb1[(16 + l16) * 20 + k0 + koff];
    b.y = Bb1[(16 + l16) * 20 + k0 + koff + 1];
    c4b = __builtin_amdgcn_wmma_f32_16x16x4_f32(false, a, false, b, (short)0,
                                                c4b, false, false);
    b.x = Bb1[(32 + l16) * 20 + k0 + koff];
    b.y = Bb1[(32 + l16) * 20 + k0 + koff + 1];
    c4c = __builtin_amdgcn_wmma_f32_16x16x4_f32(false, a, false, b, (short)0,
                                                c4c, false, false);
  }
  // scatter dE rows into sorted order
#pragma unroll
  for (int v = 0; v < 8; ++v) {
    const int m = v + mof;
    const int prow = pos[atomBase + m];
    float* __restrict__ drow = dE + (size_t)prow * PP;
    drow[l16] = c4a[v];
    drow[16 + l16] = c4b[v];
    const int n3 = 32 + l16;
    if (n3 < PP) drow[n3] = c4c[v];
  }
}

// =====================================================================
// Deterministic per-image energy reduction over sorted segments
// =====================================================================
__global__ __launch_bounds__(128) void energy_kernel(
    const int* __restrict__ start, const float* __restrict__ atomE,
    float* __restrict__ out) {
  __shared__ float red[128];
  const int b = blockIdx.x;
  const int s0 = start[b];
  const int s1 = start[b + 1];
  float acc = 0.0f;
  for (int k = s0 + threadIdx.x; k < s1; k += 128) acc += atomE[k];
  red[threadIdx.x] = acc;
  __syncthreads();
  for (int o = 64; o > 0; o >>= 1) {
    if (threadIdx.x < o) red[threadIdx.x] += red[threadIdx.x + o];
    __syncthreads();
  }
  if (threadIdx.x == 0) out[b] = red[0];
}

// =====================================================================
// Output zeroing (d_out is poisoned before timing)
// =====================================================================
__global__ __launch_bounds__(256) void zero_kernel(float* __restrict__ p, int n) {
  const int i = blockIdx.x * 256 + threadIdx.x;
  if (i < n) p[i] = 0.0f;
}

// =====================================================================
// Sparse COO mat-vec: force[col] -= val * dE_sorted[row]
// Vectorized x4 streaming; dE_sorted (28.8MB) lives in L2.
// =====================================================================
__global__ __launch_bounds__(256) void force_kernel4(
    const int4* __restrict__ rows4, const int4* __restrict__ cols4,
    const float4* __restrict__ vals4, const float* __restrict__ dE,
    float* __restrict__ force, int nq) {
  const int i = blockIdx.x * 256 + threadIdx.x;
  if (i < nq) {
    const int4 r = rows4[i];
    const int4 c = cols4[i];
    const float4 v = vals4[i];
    unsafeAtomicAdd(&force[c.x], -v.x * dE[r.x]);
    unsafeAtomicAdd(&force[c.y], -v.y * dE[r.y]);
    unsafeAtomicAdd(&force[c.z], -v.z * dE[r.z]);
    unsafeAtomicAdd(&force[c.w], -v.w * dE[r.w]);
  }
}

__global__ __launch_bounds__(256) void force_kernel_tail(
    const int* __restrict__ rows, const int* __restrict__ cols,
    const float* __restrict__ vals, const float* __restrict__ dE,
    float* __restrict__ force, int base, int nnz) {
  const int i = base + blockIdx.x * 256 + threadIdx.x;
  if (i < nnz) unsafeAtomicAdd(&force[cols[i]], -vals[i] * dE[rows[i]]);
}

// =====================================================================
extern "C" void kernel_launch(void* const* d_in, const int* in_sizes, int n_in,
                              void* d_out, int out_size, void* d_ws,
                              size_t ws_size, hipStream_t stream) {
  const float* fps   = (const float*)d_in[0];
  const float* W1    = (const float*)d_in[1];
  const float* b1    = (const float*)d_in[2];
  const float* W2    = (const float*)d_in[3];
  const float* b2    = (const float*)d_in[4];
  const float* W3    = (const float*)d_in[5];
  const float* b3    = (const float*)d_in[6];
  const int*   imgIdx = (const int*)d_in[7];
  const int*   rows  = (const int*)d_in[8];
  const int*   cols  = (const int*)d_in[9];
  const float* vals  = (const float*)d_in[10];
  const int nnz = in_sizes[8];

  char* ws = (char*)d_ws;
  size_t off = 0;
  int* chunkHist = (int*)(ws + off); off += (size_t)NCHUNK * NB * 4;
  int* counts    = (int*)(ws + off); off += (size_t)NB * 4;
  int* start     = (int*)(ws + off); off += (size_t)(NB + 1) * 4;
  off = (off + 255) & ~(size_t)255;
  int*   pos   = (int*)(ws + off);   off += (size_t)Qc * 4;
  float* atomE = (float*)(ws + off); off += (size_t)Qc * 4;
  float* dE    = (float*)(ws + off); off += (size_t)Qc * PP * 4;

  float* energy = (float*)d_out;
  float* force  = energy + NIMG;

  // 1) deterministic stable counting sort of image_idx -> pos[]
  chunk_hist_kernel<<<NCHUNK, 256, 0, stream>>>(imgIdx, chunkHist);
  col_prefix_kernel<<<(NB + 255) / 256, 256, 0, stream>>>(chunkHist, counts);
  scan_kernel<<<1, 256, 0, stream>>>(counts, start);
  rank_kernel<<<NCHUNK, 256, 0, stream>>>(imgIdx, chunkHist, start, pos);

  // 2) zero outputs
  zero_kernel<<<(out_size + 255) / 256, 256, 0, stream>>>((float*)d_out,
                                                          out_size);

  // 3) WMMA MLP fwd+bwd; writes sorted atom energies + sorted jacobian
  mlp_kernel<<<Qc / 32, 64, 0, stream>>>(fps, W1, b1, W2, b2, W3, b3, pos,
                                         atomE, dE);

  // 4) deterministic per-image energy sums
  energy_kernel<<<NIMG, 128, 0, stream>>>(start, atomE, energy);

  // 5) sparse force accumulation (x4 vectorized + tail)
  const int nq = nnz / 4;
  if (nq > 0)
    force_kernel4<<<(nq + 255) / 256, 256, 0, stream>>>(
        (const int4*)rows, (const int4*)cols, (const float4*)vals, dE, force,
        nq);
  const int rem = nnz - nq * 4;
  if (rem > 0)
    force_kernel_tail<<<1, 256, 0, stream>>>(rows, cols, vals, dE, force,
                                             nq * 4, nnz);
}